// LSTM_79465484910828
// MI455X (gfx1250) — compile-verified
//
#include <hip/hip_runtime.h>
#include <math.h>

// LSTM forward on MI455X (gfx1250), fp32 via V_WMMA_F32_16X16X4_F32.
// S=512, B=64, D=512, H=1024, gates G=4H=4096. Gate order: i, f, j, o.
//
// Per step: grid = H/16 = 64 workgroups (one 16-col slice of H each).
// Block = 1024 threads = 32 waves = (kq in 0..7) x (gate in 0..3).
// Each wave owns all 4 M-tiles (B=64 rows) of one gate's 16-col slice and a
// 1/8 slice of K (K=192 of 1536): one B fragment feeds 4 consecutive WMMAs,
// cutting vmem/wmma from 3 to 1.5.

#define S_LEN 512
#define B_DIM 64
#define D_DIM 512
#define H_DIM 1024
#define G_DIM (4 * H_DIM)
#define FORGET_BIAS 1.0f
#define PAD 20  // 8 rows * 20 floats = 160 -> +32 banks mod 64: conflict-free spill

typedef float v2f __attribute__((ext_vector_type(2)));
typedef float v8f __attribute__((ext_vector_type(8)));

__device__ __forceinline__ v8f wmma4_f32(v2f a, v2f b, v8f c) {
    // D = A(16x4,f32) * B(4x16,f32) + C(16x16,f32)
    return __builtin_amdgcn_wmma_f32_16x16x4_f32(
        /*neg_a=*/false, a, /*neg_b=*/false, b,
        /*c_mod=*/(short)0, c, /*reuse_a=*/false, /*reuse_b=*/false);
}

__device__ __forceinline__ float sigmoidf_dev(float x) {
    return 1.0f / (1.0f + __expf(-x));
}

// Accumulate K steps (K % 4 == 0) into 4 M-tiles.
//   Arow: A + laneLo*LDA + laneHi*2 + k_offset   (row-major A, leading dim LDA)
//   Bc0 : B + (laneHi*2 + k_offset)*G_DIM + col  (row-major B, leading dim G_DIM)
// One B fragment (2 x b32) is reused by 4 WMMAs (one per M-tile).
template <int K, int LDA>
__device__ __forceinline__ void gemm4(const float* __restrict__ Arow,
                                      const float* __restrict__ Bc0,
                                      v8f (&acc)[4]) {
    const float* Bc1 = Bc0 + G_DIM;
#pragma unroll 4
    for (int k = 0; k < K; k += 4) {
        v2f b;
        b[0] = Bc0[(size_t)k * G_DIM];
        b[1] = Bc1[(size_t)k * G_DIM];
#pragma unroll
        for (int m = 0; m < 4; ++m) {
            v2f a = *(const v2f*)(Arow + (size_t)m * 16 * LDA + k);
            acc[m] = wmma4_f32(a, b, acc[m]);
        }
    }
}

__global__ __launch_bounds__(1024, 1) void lstm_step_kernel(
    const float* __restrict__ x_t,   // (B, D) row-major
    const float* __restrict__ h_in,  // (B, H)
    const float* __restrict__ c_in,  // (B, H)
    const float* __restrict__ Wx,    // (D, 4H)
    const float* __restrict__ Wh,    // (H, 4H)
    const float* __restrict__ bh,    // (4H)
    float* __restrict__ h_out,       // (B, H)
    float* __restrict__ c_out) {     // (B, H)
    __shared__ float part[8][4][B_DIM][PAD];  // 160 KB (<= 320 KB/WG on CDNA5)

    const int tid    = threadIdx.x;
    const int lane   = tid & 31;
    const int wave   = tid >> 5;
    const int gate   = wave & 3;   // 0=i 1=f 2=j 3=o
    const int kq     = wave >> 2;  // K-split octant, 0..7
    const int laneLo = lane & 15;  // N (B-frag col) / M (A-frag row) index
    const int laneHi = lane >> 4;  // selects K pair {0,1} vs {2,3}

    const int n0   = blockIdx.x << 4;            // H-column slice base
    const int bcol = gate * H_DIM + n0 + laneLo; // column in (., 4H) weights

    v8f acc[4] = {};

    // Per-lane fragment base pointers.
    const float* Bx = Wx + (size_t)(laneHi * 2) * G_DIM + bcol;
    const float* Bh = Wh + (size_t)(laneHi * 2) * G_DIM + bcol;
    const float* xr = x_t + (size_t)laneLo * D_DIM + laneHi * 2;
    const float* hr = h_in + (size_t)laneLo * H_DIM + laneHi * 2;

    // K=1536 total (x:512 then h:1024), split into 8 x 192. Wave-uniform branch.
    if (kq == 0) {
        gemm4<192, D_DIM>(xr, Bx, acc);
    } else if (kq == 1) {
        gemm4<192, D_DIM>(xr + 192, Bx + (size_t)192 * G_DIM, acc);
    } else if (kq == 2) {
        gemm4<128, D_DIM>(xr + 384, Bx + (size_t)384 * G_DIM, acc);
        gemm4<64, H_DIM>(hr, Bh, acc);
    } else if (kq == 3) {
        gemm4<192, H_DIM>(hr + 64, Bh + (size_t)64 * G_DIM, acc);
    } else {
        const int o = 256 + (kq - 4) * 192;  // 256, 448, 640, 832
        gemm4<192, H_DIM>(hr + o, Bh + (size_t)o * G_DIM, acc);
    }

    // Spill C tiles: VGPR r of tile m holds row (m*16 + laneHi*8 + r), col laneLo.
#pragma unroll
    for (int m = 0; m < 4; ++m) {
#pragma unroll
        for (int r = 0; r < 8; ++r) {
            const int row = (m << 4) + (laneHi << 3) + r;
            part[kq][gate][row][laneLo] = acc[m][r];
        }
    }
    __syncthreads();

    // Elementwise LSTM cell update: 1024 threads, one (row, col) each.
    const int row  = tid >> 4;
    const int col  = tid & 15;
    const int hIdx = n0 + col;
    const size_t off = (size_t)row * H_DIM + hIdx;

    float gi = bh[hIdx];
    float gf = bh[H_DIM + hIdx];
    float gj = bh[2 * H_DIM + hIdx];
    float go = bh[3 * H_DIM + hIdx];
#pragma unroll
    for (int q = 0; q < 8; ++q) {
        gi += part[q][0][row][col];
        gf += part[q][1][row][col];
        gj += part[q][2][row][col];
        go += part[q][3][row][col];
    }

    const float cprev = c_in[off];
    const float cnew  = cprev * sigmoidf_dev(gf + FORGET_BIAS) +
                        sigmoidf_dev(gi) * tanhf(gj);
    const float hnew  = tanhf(cnew) * sigmoidf_dev(go);

    c_out[off] = cnew;
    h_out[off] = hnew;
}

extern "C" void kernel_launch(void* const* d_in, const int* in_sizes, int n_in,
                              void* d_out, int out_size, void* d_ws, size_t ws_size,
                              hipStream_t stream) {
    const float* x  = (const float*)d_in[0];  // (S, B, D)
    const float* h0 = (const float*)d_in[1];  // (B, H)
    const float* c0 = (const float*)d_in[2];  // (B, H)
    const float* Wx = (const float*)d_in[3];  // (D, 4H)
    const float* Wh = (const float*)d_in[4];  // (H, 4H)
    const float* bh = (const float*)d_in[5];  // (4H)

    float* out_h = (float*)d_out;                 // (B, H)
    float* out_c = out_h + (size_t)B_DIM * H_DIM; // (B, H)

    // Workspace: double-buffered h + single c buffer (each 256 KB).
    float* hbuf0 = (float*)d_ws;
    float* hbuf1 = hbuf0 + (size_t)B_DIM * H_DIM;
    float* cbuf  = hbuf1 + (size_t)B_DIM * H_DIM;

    const dim3 grid(H_DIM / 16);
    const dim3 block(1024);

    for (int s = 0; s < S_LEN; ++s) {
        const float* x_t  = x + (size_t)s * B_DIM * D_DIM;
        const float* h_in = (s == 0) ? h0 : ((s & 1) ? hbuf1 : hbuf0);
        const float* c_in = (s == 0) ? c0 : cbuf;
        float* h_out = (s == S_LEN - 1) ? out_h : (((s + 1) & 1) ? hbuf1 : hbuf0);
        float* c_out = (s == S_LEN - 1) ? out_c : cbuf;

        lstm_step_kernel<<<grid, block, 0, stream>>>(x_t, h_in, c_in, Wx, Wh,
                                                     bh, h_out, c_out);
    }
}